// S3ANet_52518860096117
// MI455X (gfx1250) — compile-verified
//
#include <hip/hip_runtime.h>
#include <math.h>

typedef __attribute__((ext_vector_type(2))) float v2f;
typedef __attribute__((ext_vector_type(8))) float v8f;

// fp32 dense WMMA: D = A(16x4) * B(4x16) + C(16x16), full fp32 semantics.
#define WMMA_F32X4(a, b, c) \
  __builtin_amdgcn_wmma_f32_16x16x4_f32(false, (a), false, (b), (short)0, (c), false, false)

__device__ __forceinline__ float act_apply(float v, int act) {
  if (act == 1) return v > 0.f ? v : 0.f;
  if (act == 2) return 0.5f * v * (1.f + erff(v * 0.70710678118654752f)); // exact gelu
  return v;
}

// ---------------------------------------------------------------------------
// Pack OIHW conv weights into WMMA-A-fragment order, zero-padded over K:
//   wpk[(((ot*9 + tap)*nch + chunk)*32 + lane)*2 + r] = W[ot*16+lane%16, chunk*4+2*(lane/16)+r, tap]
// One contiguous 8B load per lane per (tap,chunk) in the conv kernel; no guards.
// ---------------------------------------------------------------------------
__global__ void pack_conv_w(const float* __restrict__ w, float* __restrict__ wpk,
                            int OC, int IC) {
  const int nch = (IC + 3) >> 2;
  const long n = (long)(OC >> 4) * 9 * nch * 32;
  long i = (long)blockIdx.x * blockDim.x + threadIdx.x;
  if (i >= n) return;
  const int lane = (int)(i & 31); long t = i >> 5;
  const int chunk = (int)(t % nch); t /= nch;
  const int tap = (int)(t % 9); t /= 9;
  const int ot = (int)t;
  const int col = lane & 15, khalf = lane >> 4;
  const int m = ot * 16 + col;
  const int k0 = chunk * 4 + 2 * khalf;
  const size_t base = (size_t)m * IC * 9 + tap;   // W[m,k,tap] = m*IC*9 + k*9 + tap
  const float a0 = (k0     < IC) ? w[base + (size_t)k0 * 9]       : 0.f;
  const float a1 = (k0 + 1 < IC) ? w[base + (size_t)(k0 + 1) * 9] : 0.f;
  float* o = wpk + (((size_t)(ot * 9 + tap) * nch + chunk) * 32 + lane) * 2;
  o[0] = a0; o[1] = a1;
}

// ---------------------------------------------------------------------------
// 3x3 valid conv (dilated) as implicit GEMM on V_WMMA_F32_16X16X4_F32.
// One wave computes a 16(oc) x 32(pixel) tile: one shared A fragment feeds two
// B fragments / accumulators per K-chunk (2 WMMAs per A load). Branch-free
// main K loop; single guarded tail chunk when IC%4!=0 (conv0 only).
// Fragment layouts per CDNA5 ISA 7.12.2:
//   A[m][k]: lane = m + 16*(k>=2), vgpr = k&1     B[k][n]: lane = n + 16*(k>=2), vgpr = k&1
//   D[m][n]: lane = n + 16*(m>=8), vgpr = m&7
// ---------------------------------------------------------------------------
__global__ __launch_bounds__(256)
void conv3x3_wmma_relu(const float* __restrict__ in, const float* __restrict__ wpk,
                       const float* __restrict__ bias, float* __restrict__ out,
                       int Bn, int IC, int Hin, int Win, int OC, int Hout, int Wout, int dil) {
  const int P = Hout * Wout;
  const int ntiles = P >> 4;               // 16-pixel tiles (P always multiple of 16 here)
  const int pt2n = (ntiles + 1) >> 1;      // tile pairs
  const int octiles = OC >> 4;
  const long total = (long)Bn * pt2n * octiles;
  const long wid = (long)blockIdx.x * (blockDim.x >> 5) + (threadIdx.x >> 5);
  if (wid >= total) return;
  const int lane = threadIdx.x & 31;
  const int col = lane & 15;
  const int khalf = lane >> 4;

  long t = wid;
  const int ot  = (int)(t % octiles); t /= octiles;
  const int pt2 = (int)(t % pt2n);    t /= pt2n;
  const int b   = (int)t;

  const bool has2 = (2 * pt2 + 1) < ntiles;      // wave-uniform
  const int p0 = pt2 * 32 + col;
  const int p1 = has2 ? p0 + 16 : p0;
  const int y0 = p0 / Wout, x0 = p0 % Wout;
  const int y1 = p1 / Wout, x1 = p1 % Wout;

  const int nch = (IC + 3) >> 2;
  const int cmain = IC >> 2;
  const size_t inHW = (size_t)Hin * Win;
  const float* inb = in + (size_t)b * IC * inHW;
  const float* wk = wpk + (size_t)(ot * 9) * nch * 64 + lane * 2;
  const int kbase = 2 * khalf;

  v8f acc0 = {0.f, 0.f, 0.f, 0.f, 0.f, 0.f, 0.f, 0.f};
  v8f acc1 = {0.f, 0.f, 0.f, 0.f, 0.f, 0.f, 0.f, 0.f};

#pragma unroll
  for (int kh = 0; kh < 3; ++kh)
#pragma unroll
    for (int kw = 0; kw < 3; ++kw) {
      const int tap = kh * 3 + kw;
      const float* ip0 = inb + (size_t)(y0 + kh * dil) * Win + (x0 + kw * dil);
      const float* ip1 = inb + (size_t)(y1 + kh * dil) * Win + (x1 + kw * dil);
      const float* wkt = wk + (size_t)tap * nch * 64;
#pragma unroll 2
      for (int c = 0; c < cmain; ++c) {
        const v2f a = *(const v2f*)(wkt + (size_t)c * 64);
        const size_t k0 = (size_t)(c * 4 + kbase);
        v2f b0, b1;
        b0.x = ip0[k0 * inHW]; b0.y = ip0[(k0 + 1) * inHW];
        b1.x = ip1[k0 * inHW]; b1.y = ip1[(k0 + 1) * inHW];
        acc0 = WMMA_F32X4(a, b0, acc0);
        acc1 = WMMA_F32X4(a, b1, acc1);
      }
      if (cmain < nch) {                       // K tail (weights already zero-padded)
        const v2f a = *(const v2f*)(wkt + (size_t)cmain * 64);
        const int k0 = cmain * 4 + kbase;
        v2f b0 = {0.f, 0.f}, b1 = {0.f, 0.f};
        if (k0 < IC)     { b0.x = ip0[(size_t)k0 * inHW];       b1.x = ip1[(size_t)k0 * inHW]; }
        if (k0 + 1 < IC) { b0.y = ip0[(size_t)(k0 + 1) * inHW]; b1.y = ip1[(size_t)(k0 + 1) * inHW]; }
        acc0 = WMMA_F32X4(a, b0, acc0);
        acc1 = WMMA_F32X4(a, b1, acc1);
      }
    }

  float* ob = out + (size_t)b * OC * P;
#pragma unroll
  for (int r = 0; r < 8; ++r) {
    const int mm = ot * 16 + r + 8 * khalf;
    float v = acc0[r] + bias[mm];
    ob[(size_t)mm * P + p0] = v > 0.f ? v : 0.f;
  }
  if (has2) {
#pragma unroll
    for (int r = 0; r < 8; ++r) {
      const int mm = ot * 16 + r + 8 * khalf;
      float v = acc1[r] + bias[mm];
      ob[(size_t)mm * P + p1] = v > 0.f ? v : 0.f;
    }
  }
}

// ---------------------------------------------------------------------------
// 1x1 conv / GEMM on WMMA, 16(oc) x 32(pixel) per wave, branch-free (K%4==0 at
// every call site; A loads are contiguous v2f). Optional bias/act/residual and
// channel-sliced output (OCtot/mOff) for the 320-ch concat buffer.
// ---------------------------------------------------------------------------
__global__ __launch_bounds__(256)
void gemm1x1_wmma(const float* __restrict__ in, const float* __restrict__ w,
                  const float* __restrict__ bias, const float* __restrict__ resid,
                  float* __restrict__ out,
                  int Bn, int K, int ICtot, int M, int OCtot, int mOff, int P, int act) {
  const int ptiles = P >> 4;
  const int pt2n = (ptiles + 1) >> 1;
  const int mtiles = M >> 4;
  const long total = (long)Bn * pt2n * mtiles;
  const long wid = (long)blockIdx.x * (blockDim.x >> 5) + (threadIdx.x >> 5);
  if (wid >= total) return;
  const int lane = threadIdx.x & 31;
  const int col = lane & 15;
  const int khalf = lane >> 4;
  long t = wid;
  const int ot  = (int)(t % mtiles); t /= mtiles;
  const int pt2 = (int)(t % pt2n);   t /= pt2n;
  const int b = (int)t;
  const bool has2 = (2 * pt2 + 1) < ptiles;
  const int p0 = pt2 * 32 + col;
  const int p1 = has2 ? p0 + 16 : p0;
  const int m = ot * 16 + col;
  const float* ib0 = in + (size_t)b * ICtot * P + p0;
  const float* ib1 = in + (size_t)b * ICtot * P + p1;
  const float* wp = w + (size_t)m * K;
  v8f acc0 = {0.f, 0.f, 0.f, 0.f, 0.f, 0.f, 0.f, 0.f};
  v8f acc1 = {0.f, 0.f, 0.f, 0.f, 0.f, 0.f, 0.f, 0.f};
#pragma unroll 2
  for (int k0i = 0; k0i < K; k0i += 4) {
    const size_t k0 = (size_t)(k0i + 2 * khalf);
    const v2f a = *(const v2f*)(wp + k0);
    v2f b0, b1;
    b0.x = ib0[k0 * P]; b0.y = ib0[(k0 + 1) * P];
    b1.x = ib1[k0 * P]; b1.y = ib1[(k0 + 1) * P];
    acc0 = WMMA_F32X4(a, b0, acc0);
    acc1 = WMMA_F32X4(a, b1, acc1);
  }
  float* ob = out + ((size_t)b * OCtot + mOff) * P;
#pragma unroll
  for (int r = 0; r < 8; ++r) {
    const int mm = ot * 16 + r + 8 * khalf;
    float v = acc0[r];
    if (bias) v += bias[mm];
    v = act_apply(v, act);
    if (resid) v += resid[((size_t)b * M + mm) * P + p0];
    ob[(size_t)mm * P + p0] = v;
  }
  if (has2) {
#pragma unroll
    for (int r = 0; r < 8; ++r) {
      const int mm = ot * 16 + r + 8 * khalf;
      float v = acc1[r];
      if (bias) v += bias[mm];
      v = act_apply(v, act);
      if (resid) v += resid[((size_t)b * M + mm) * P + p1];
      ob[(size_t)mm * P + p1] = v;
    }
  }
}

// ------------------------- elementwise / small kernels ----------------------
__global__ void avgpool2x2(const float* __restrict__ in, float* __restrict__ out,
                           long n, int Hin, int Win) {
  long i = (long)blockIdx.x * blockDim.x + threadIdx.x;
  if (i >= n) return;
  const int Wout = Win >> 1, Hout = Hin >> 1;
  int x = (int)(i % Wout); long t = i / Wout;
  int y = (int)(t % Hout); long bc = t / Hout;
  const float* ip = in + bc * (size_t)Hin * Win + (size_t)(2 * y) * Win + 2 * x;
  out[i] = 0.25f * (ip[0] + ip[1] + ip[Win] + ip[Win + 1]);
}

__global__ void copy_ch(const float* __restrict__ in, float* __restrict__ out,
                        int Bn, int C, int OCtot, int off, int P) {
  long n = (long)Bn * C * P;
  long i = (long)blockIdx.x * blockDim.x + threadIdx.x;
  if (i >= n) return;
  int p = (int)(i % P); long t = i / P;
  int c = (int)(t % C); int b = (int)(t / C);
  out[((size_t)b * OCtot + off + c) * P + p] = in[i];
}

// per-pixel L2 normalize over C channels: v *= 1/(||v||+eps)
__global__ void pixel_l2norm(float* __restrict__ t, int Bn, int C, int P, float eps) {
  long n = (long)Bn * P;
  long i = (long)blockIdx.x * blockDim.x + threadIdx.x;
  if (i >= n) return;
  int p = (int)(i % P); int b = (int)(i / P);
  float* base = t + (size_t)b * C * P + p;
  float s = 0.f;
  for (int c = 0; c < C; ++c) { float v = base[(size_t)c * P]; s += v * v; }
  float inv = 1.f / (sqrtf(s) + eps);
  for (int c = 0; c < C; ++c) base[(size_t)c * P] *= inv;
}

// criss-cross scores per pixel within an SxS window of the 24x24 map; softmax over 2S.
template <int S>
__global__ void cc_scores(const float* __restrict__ q, const float* __restrict__ k,
                          float* __restrict__ sc, int Bn) {
  long n = (long)Bn * 576;
  long idx = (long)blockIdx.x * blockDim.x + threadIdx.x;
  if (idx >= n) return;
  const int p = (int)(idx % 576);
  const int b = (int)(idx / 576);
  const int i = p / 24, j = p % 24;
  const int by = (i / S) * S, bx = (j / S) * S;
  const float* qb = q + (size_t)b * 16 * 576;
  const float* kb = k + (size_t)b * 16 * 576;
  float qv[16];
#pragma unroll
  for (int c = 0; c < 16; ++c) qv[c] = qb[(size_t)c * 576 + p];
  float sco[2 * S];
  float mx = -1e30f;
#pragma unroll
  for (int m = 0; m < S; ++m) {               // along H (same column j)
    const int pp = (by + m) * 24 + j;
    float d = 0.f;
#pragma unroll
    for (int c = 0; c < 16; ++c) d += qv[c] * kb[(size_t)c * 576 + pp];
    sco[m] = d; mx = fmaxf(mx, d);
  }
#pragma unroll
  for (int m = 0; m < S; ++m) {               // along W (same row i)
    const int pp = i * 24 + (bx + m);
    float d = 0.f;
#pragma unroll
    for (int c = 0; c < 16; ++c) d += qv[c] * kb[(size_t)c * 576 + pp];
    sco[S + m] = d; mx = fmaxf(mx, d);
  }
  float sum = 0.f;
#pragma unroll
  for (int m = 0; m < 2 * S; ++m) { float e = __expf(sco[m] - mx); sco[m] = e; sum += e; }
  const float inv = 1.f / sum;
  float* sp = sc + (size_t)idx * 48;
#pragma unroll
  for (int m = 0; m < 2 * S; ++m) sp[m] = sco[m] * inv;
}

template <int S>
__global__ void cc_apply(const float* __restrict__ sc, const float* __restrict__ v,
                         float* __restrict__ out, int Bn, int OCtot, int off) {
  long n = (long)Bn * 64 * 576;
  long idx = (long)blockIdx.x * blockDim.x + threadIdx.x;
  if (idx >= n) return;
  const int p = (int)(idx % 576); long t = idx / 576;
  const int c = (int)(t % 64);    const int b = (int)(t / 64);
  const int i = p / 24, j = p % 24;
  const int by = (i / S) * S, bx = (j / S) * S;
  const float* sp = sc + ((size_t)b * 576 + p) * 48;
  const float* vb = v + ((size_t)b * 64 + c) * 576;
  float acc = 0.f;
#pragma unroll
  for (int m = 0; m < S; ++m) acc += sp[m] * vb[(by + m) * 24 + j];
#pragma unroll
  for (int m = 0; m < S; ++m) acc += sp[S + m] * vb[i * 24 + bx + m];
  out[((size_t)b * OCtot + off + c) * 576 + p] = acc;
}

// depthwise 3x3, pad=1, on 24x24 maps
__global__ void dw3x3(const float* __restrict__ in, const float* __restrict__ w,
                      float* __restrict__ out, int Bn, int C, int act) {
  long n = (long)Bn * C * 576;
  long idx = (long)blockIdx.x * blockDim.x + threadIdx.x;
  if (idx >= n) return;
  const int p = (int)(idx % 576); long t = idx / 576;
  const int c = (int)(t % C); const int b = (int)(t / C);
  const int y = p / 24, x = p % 24;
  const float* ib = in + ((size_t)b * C + c) * 576;
  const float* wp = w + (size_t)c * 9;
  float acc = 0.f;
#pragma unroll
  for (int kh = 0; kh < 3; ++kh) {
    const int yy = y + kh - 1;
    if (yy < 0 || yy >= 24) continue;
#pragma unroll
    for (int kw = 0; kw < 3; ++kw) {
      const int xx = x + kw - 1;
      if (xx < 0 || xx >= 24) continue;
      acc += wp[kh * 3 + kw] * ib[yy * 24 + xx];
    }
  }
  out[idx] = act_apply(acc, act);
}

// group norm stats: block per (b,g); C=64, groups=16 -> 4 ch * 576 px contiguous
__global__ void gn_stats(const float* __restrict__ x, float* __restrict__ stats, int Bn) {
  const int bg = blockIdx.x;
  const float* base = x + (size_t)bg * 4 * 576;
  float s = 0.f, ss = 0.f;
  for (int e = threadIdx.x; e < 4 * 576; e += blockDim.x) {
    float v = base[e]; s += v; ss += v * v;
  }
  __shared__ float sh1[256], sh2[256];
  sh1[threadIdx.x] = s; sh2[threadIdx.x] = ss; __syncthreads();
  for (int st = 128; st > 0; st >>= 1) {
    if (threadIdx.x < st) { sh1[threadIdx.x] += sh1[threadIdx.x + st]; sh2[threadIdx.x] += sh2[threadIdx.x + st]; }
    __syncthreads();
  }
  if (threadIdx.x == 0) {
    const float m = sh1[0] * (1.f / (4 * 576));
    const float var = sh2[0] * (1.f / (4 * 576)) - m * m;
    stats[bg * 2] = m; stats[bg * 2 + 1] = rsqrtf(var + 1e-5f);
  }
}

__global__ void gn_apply_relu_add(const float* __restrict__ x, const float* __restrict__ stats,
                                  const float* __restrict__ gw, const float* __restrict__ gb,
                                  const float* __restrict__ resid, float* __restrict__ out, int Bn) {
  long n = (long)Bn * 64 * 576;
  long idx = (long)blockIdx.x * blockDim.x + threadIdx.x;
  if (idx >= n) return;
  long t = idx / 576;
  const int c = (int)(t % 64); const int b = (int)(t / 64);
  const int g = c >> 2;
  const float m = stats[(b * 16 + g) * 2];
  const float r = stats[(b * 16 + g) * 2 + 1];
  float v = (x[idx] - m) * r * gw[c] + gb[c];
  v = v > 0.f ? v : 0.f;
  out[idx] = v + resid[idx];
}

// GST: inverse L2 norms over n=576 for q/k rows (channels 0..127 of qkv2)
__global__ void row_invnorm(const float* __restrict__ x, float* __restrict__ invn, int Bn) {
  const int r = blockIdx.x;           // r = b*128 + ch
  const int b = r >> 7, ch = r & 127;
  const float* base = x + ((size_t)b * 192 + ch) * 576;
  float s = 0.f;
  for (int e = threadIdx.x; e < 576; e += blockDim.x) { float v = base[e]; s += v * v; }
  __shared__ float sh[256];
  sh[threadIdx.x] = s; __syncthreads();
  for (int st = 128; st > 0; st >>= 1) {
    if (threadIdx.x < st) sh[threadIdx.x] += sh[threadIdx.x + st];
    __syncthreads();
  }
  if (threadIdx.x == 0) invn[r] = 1.f / fmaxf(sqrtf(sh[0]), 1e-12f);
}

// GST channel attention: per (b,h) 4x4 logits + row softmax
__global__ void gst_attn(const float* __restrict__ qkv2, const float* __restrict__ invn,
                         const float* __restrict__ temp, float* __restrict__ att, int Bn) {
  const int bh = blockIdx.x;
  const int h = bh & 15, b = bh >> 4;
  __shared__ float E[16];
  const int tid = threadIdx.x;
  if (tid < 16) {
    const int c = tid >> 2, d = tid & 3;
    const float* qb = qkv2 + ((size_t)b * 192 + h * 4 + c) * 576;
    const float* kb = qkv2 + ((size_t)b * 192 + 64 + h * 4 + d) * 576;
    float s = 0.f;
    for (int n = 0; n < 576; ++n) s += qb[n] * kb[n];
    s *= invn[b * 128 + h * 4 + c] * invn[b * 128 + 64 + h * 4 + d];
    E[tid] = s * temp[h];
  }
  __syncthreads();
  if (tid < 4) {
    const int c = tid;
    float mx = fmaxf(fmaxf(E[c * 4], E[c * 4 + 1]), fmaxf(E[c * 4 + 2], E[c * 4 + 3]));
    float e[4]; float sum = 0.f;
#pragma unroll
    for (int d = 0; d < 4; ++d) { e[d] = __expf(E[c * 4 + d] - mx); sum += e[d]; }
    const float inv = 1.f / sum;
#pragma unroll
    for (int d = 0; d < 4; ++d) att[(((size_t)b * 16 + h) * 4 + c) * 4 + d] = e[d] * inv;
  }
}

__global__ void gst_o(const float* __restrict__ qkv2, const float* __restrict__ att,
                      float* __restrict__ o, int Bn) {
  long n = (long)Bn * 64 * 576;
  long idx = (long)blockIdx.x * blockDim.x + threadIdx.x;
  if (idx >= n) return;
  const int p = (int)(idx % 576); long t = idx / 576;
  const int ch = (int)(t % 64); const int b = (int)(t / 64);
  const int h = ch >> 2, c = ch & 3;
  const float* vb = qkv2 + ((size_t)b * 192 + 128 + h * 4) * 576 + p;
  const float* ab = att + (((size_t)b * 16 + h) * 4 + c) * 4;
  o[idx] = ab[0] * vb[0] + ab[1] * vb[576] + ab[2] * vb[2 * 576] + ab[3] * vb[3 * 576];
}

// LayerNorm over channel dim (NCHW, C=64) per pixel
__global__ void layernorm_ch(const float* __restrict__ x, const float* __restrict__ w,
                             const float* __restrict__ bta, float* __restrict__ out, int Bn) {
  long n = (long)Bn * 576;
  long idx = (long)blockIdx.x * blockDim.x + threadIdx.x;
  if (idx >= n) return;
  const int p = (int)(idx % 576); const int b = (int)(idx / 576);
  const float* base = x + (size_t)b * 64 * 576 + p;
  float vals[64]; float m = 0.f;
#pragma unroll
  for (int c = 0; c < 64; ++c) { vals[c] = base[(size_t)c * 576]; m += vals[c]; }
  m *= (1.f / 64.f);
  float var = 0.f;
#pragma unroll
  for (int c = 0; c < 64; ++c) { float d = vals[c] - m; var += d * d; }
  var *= (1.f / 64.f);
  const float r = rsqrtf(var + 1e-5f);
  float* ob = out + (size_t)b * 64 * 576 + p;
#pragma unroll
  for (int c = 0; c < 64; ++c) ob[(size_t)c * 576] = (vals[c] - m) * r * w[c] + bta[c];
}

// fused: bilinear(align_corners) upsample of h0/h1/x11 to 66x66 + concat + 1x1 cls
__device__ __forceinline__ void up_acc(const float* __restrict__ sb, int nin, float scale,
                                       int Y, int X, const float* __restrict__ cw, int cbase,
                                       float* acc) {
  const float sy = Y * scale, sx = X * scale;
  int y0 = (int)sy; if (y0 > nin - 1) y0 = nin - 1;
  int x0 = (int)sx; if (x0 > nin - 1) x0 = nin - 1;
  const int y1 = (y0 + 1 > nin - 1) ? nin - 1 : y0 + 1;
  const int x1 = (x0 + 1 > nin - 1) ? nin - 1 : x0 + 1;
  const float wy = sy - (float)y0, wx = sx - (float)x0;
  for (int c = 0; c < 64; ++c) {
    const float* cp = sb + (size_t)c * nin * nin;
    const float v00 = cp[y0 * nin + x0], v01 = cp[y0 * nin + x1];
    const float v10 = cp[y1 * nin + x0], v11 = cp[y1 * nin + x1];
    const float g = (1.f - wy) * ((1.f - wx) * v00 + wx * v01)
                  + wy * ((1.f - wx) * v10 + wx * v11);
#pragma unroll
    for (int o = 0; o < 9; ++o) acc[o] += cw[o * 192 + cbase + c] * g;
  }
}

__global__ void final_cls(const float* __restrict__ h0, const float* __restrict__ h1,
                          const float* __restrict__ x11, const float* __restrict__ cw,
                          const float* __restrict__ cb, float* __restrict__ out, int Bn) {
  long n = (long)Bn * 66 * 66;
  long idx = (long)blockIdx.x * blockDim.x + threadIdx.x;
  if (idx >= n) return;
  const int X = (int)(idx % 66); long t = idx / 66;
  const int Y = (int)(t % 66); const int b = (int)(t / 66);
  float acc[9];
#pragma unroll
  for (int o = 0; o < 9; ++o) acc[o] = cb[o];
  up_acc(h0  + (size_t)b * 64 * 64 * 64, 64, 63.f / 65.f, Y, X, cw, 0,   acc);
  up_acc(h1  + (size_t)b * 64 * 60 * 60, 60, 59.f / 65.f, Y, X, cw, 64,  acc);
  up_acc(x11 + (size_t)b * 64 * 24 * 24, 24, 23.f / 65.f, Y, X, cw, 128, acc);
#pragma unroll
  for (int o = 0; o < 9; ++o)
    out[(((size_t)b * 9 + o) * 66 + Y) * 66 + X] = acc[o];
}

// ---------------------------------------------------------------------------
static inline int wgrid(long waves) { return (int)((waves + 7) / 8); }  // 256 thr = 8 waves
static inline int tgrid(long n) { return (int)((n + 255) / 256); }
static inline long cpairs(int ntiles) { return (ntiles + 1) / 2; }

static void launch_cc(const float* q, const float* k, const float* v, float* sc,
                      float* dst, int OCtot, int off, int s, int B, hipStream_t stream) {
  const long n1 = (long)B * 576, n2 = (long)B * 64 * 576;
  switch (s) {
    case 24: cc_scores<24><<<tgrid(n1), 256, 0, stream>>>(q, k, sc, B);
             cc_apply<24><<<tgrid(n2), 256, 0, stream>>>(sc, v, dst, B, OCtot, off); break;
    case 12: cc_scores<12><<<tgrid(n1), 256, 0, stream>>>(q, k, sc, B);
             cc_apply<12><<<tgrid(n2), 256, 0, stream>>>(sc, v, dst, B, OCtot, off); break;
    case 8:  cc_scores<8><<<tgrid(n1), 256, 0, stream>>>(q, k, sc, B);
             cc_apply<8><<<tgrid(n2), 256, 0, stream>>>(sc, v, dst, B, OCtot, off); break;
    default: cc_scores<4><<<tgrid(n1), 256, 0, stream>>>(q, k, sc, B);
             cc_apply<4><<<tgrid(n2), 256, 0, stream>>>(sc, v, dst, B, OCtot, off); break;
  }
}

extern "C" void kernel_launch(void* const* d_in, const int* in_sizes, int n_in,
                              void* d_out, int out_size, void* d_ws, size_t ws_size,
                              hipStream_t stream) {
  (void)in_sizes; (void)n_in; (void)out_size; (void)ws_size;
  const float* x          = (const float*)d_in[0];
  const float* conv0_w    = (const float*)d_in[1];
  const float* conv0_b    = (const float*)d_in[2];
  const float* conv1_w    = (const float*)d_in[3];
  const float* conv1_b    = (const float*)d_in[4];
  const float* conv2_w    = (const float*)d_in[5];
  const float* conv2_b    = (const float*)d_in[6];
  const float* head_red_w = (const float*)d_in[7];
  const float* pa_qw      = (const float*)d_in[8];
  const float* pa_qb      = (const float*)d_in[9];
  const float* pa_kw      = (const float*)d_in[10];
  const float* pa_kb      = (const float*)d_in[11];
  const float* pa_vw      = (const float*)d_in[12];
  const float* pa_vb      = (const float*)d_in[13];
  const float* head_out_w = (const float*)d_in[14];
  const float* gn_w       = (const float*)d_in[15];
  const float* gn_b       = (const float*)d_in[16];
  const float* gst_temp   = (const float*)d_in[17];
  const float* gst_qkv_w  = (const float*)d_in[18];
  const float* gst_dw_w   = (const float*)d_in[19];
  const float* gst_proj_w = (const float*)d_in[20];
  const float* ln_w       = (const float*)d_in[21];
  const float* ln_b       = (const float*)d_in[22];
  const float* ff_w1      = (const float*)d_in[23];
  const float* ff_dw_w    = (const float*)d_in[24];
  const float* ff_w2      = (const float*)d_in[25];
  const float* cls_w      = (const float*)d_in[26];
  const float* cls_b      = (const float*)d_in[27];

  const int B = 64;
  float* ws = (float*)d_ws;
  size_t off_ = 0;
  auto alloc = [&](size_t nf) { float* p = ws + off_; off_ += nf; return p; };
  float* h0   = alloc(16777216);  // B*64*64*64
  float* h1   = alloc(14745600);  // B*64*60*60
  float* h2   = alloc(2359296);   // B*64*576
  float* r3   = alloc(9437184);   // pool / qkv / gst_o / ffn-mid(256ch)
  float* r4   = alloc(11796480);  // feats(320ch) / qkv2 / ffn-mid2
  float* xr   = alloc(2359296);
  float* qb   = alloc(589824);    // B*16*576
  float* kb   = alloc(589824);
  float* vb   = alloc(2359296);
  float* t1   = alloc(2359296);   // fucont ping / head_out pre-GN
  float* xh   = alloc(2359296);   // xh, later reused as LN output
  float* xg   = alloc(2359296);
  float* x11  = alloc(2359296);
  float* sc   = alloc(1769472);   // B*576*48 CC softmax weights
  float* stats= alloc(2048);      // GN mean/rstd
  float* invn = alloc(8192);      // GST row inv-norms
  float* att  = alloc(16384);     // GST 4x4 attn per (b,h)
  float* wpk0 = alloc(59904);     // 4*9*26*64 packed conv0 weights
  float* wpk1 = alloc(36864);     // 4*9*16*64 packed conv1 weights
  float* wpk2 = alloc(36864);     // 4*9*16*64 packed conv2 weights

  // 0) pack conv weights into WMMA fragment order (zero-padded K)
  pack_conv_w<<<tgrid(4L * 9 * 26 * 32), 256, 0, stream>>>(conv0_w, wpk0, 64, 103);
  pack_conv_w<<<tgrid(4L * 9 * 16 * 32), 256, 0, stream>>>(conv1_w, wpk1, 64, 64);
  pack_conv_w<<<tgrid(4L * 9 * 16 * 32), 256, 0, stream>>>(conv2_w, wpk2, 64, 64);

  // 1) conv0 103->64, 66x66 -> 64x64, relu
  conv3x3_wmma_relu<<<wgrid((long)B * cpairs(256) * 4), 256, 0, stream>>>(
      x, wpk0, conv0_b, h0, B, 103, 66, 66, 64, 64, 64, 1);
  // 2) conv1 dil=2, 64x64 -> 60x60, relu
  conv3x3_wmma_relu<<<wgrid((long)B * cpairs(225) * 4), 256, 0, stream>>>(
      h0, wpk1, conv1_b, h1, B, 64, 64, 64, 64, 60, 60, 2);
  // 3) avgpool 2x2 -> 30x30
  avgpool2x2<<<tgrid((long)B * 64 * 900), 256, 0, stream>>>(h1, r3, (long)B * 64 * 900, 60, 60);
  // 4) conv2 dil=3, 30x30 -> 24x24, relu -> h2
  conv3x3_wmma_relu<<<wgrid((long)B * cpairs(36) * 4), 256, 0, stream>>>(
      r3, wpk2, conv2_b, h2, B, 64, 30, 30, 64, 24, 24, 3);
  // 5) xr = head_red(h2)
  gemm1x1_wmma<<<wgrid((long)B * 18 * 4), 256, 0, stream>>>(
      h2, head_red_w, nullptr, nullptr, xr, B, 64, 64, 64, 64, 0, 576, 0);
  // 6) feats[:,0:64] = h2
  copy_ch<<<tgrid((long)B * 64 * 576), 256, 0, stream>>>(h2, r4, B, 64, 320, 0, 576);
  // 7) 4 bin levels x 2 iterations of grid criss-cross attention
  const int binsArr[4] = {1, 2, 3, 6};
  for (int i = 0; i < 4; ++i) {
    const int s = 24 / binsArr[i];
    const float* qw = pa_qw + (size_t)i * 16 * 64; const float* qbias = pa_qb + (size_t)i * 16;
    const float* kw = pa_kw + (size_t)i * 16 * 64; const float* kbias = pa_kb + (size_t)i * 16;
    const float* vw = pa_vw + (size_t)i * 64 * 64; const float* vbias = pa_vb + (size_t)i * 64;
    const float* tin = xr;
    for (int it = 0; it < 2; ++it) {
      gemm1x1_wmma<<<wgrid((long)B * 18 * 1), 256, 0, stream>>>(
          tin, qw, qbias, nullptr, qb, B, 64, 64, 16, 16, 0, 576, 0);
      gemm1x1_wmma<<<wgrid((long)B * 18 * 1), 256, 0, stream>>>(
          tin, kw, kbias, nullptr, kb, B, 64, 64, 16, 16, 0, 576, 0);
      gemm1x1_wmma<<<wgrid((long)B * 18 * 4), 256, 0, stream>>>(
          tin, vw, vbias, nullptr, vb, B, 64, 64, 64, 64, 0, 576, 0);
      pixel_l2norm<<<tgrid((long)B * 576), 256, 0, stream>>>(qb, B, 16, 576, 1e-10f);
      pixel_l2norm<<<tgrid((long)B * 576), 256, 0, stream>>>(kb, B, 16, 576, 1e-10f);
      if (it == 0) launch_cc(qb, kb, vb, sc, t1, 64, 0, s, B, stream);
      else         launch_cc(qb, kb, vb, sc, r4, 320, 64 + 64 * i, s, B, stream);
      tin = t1;
    }
  }
  // 8) head_out 320->64 (pre-GN) into t1
  gemm1x1_wmma<<<wgrid((long)B * 18 * 4), 256, 0, stream>>>(
      r4, head_out_w, nullptr, nullptr, t1, B, 320, 320, 64, 64, 0, 576, 0);
  // 9) GN(16 groups) + relu + residual h2 -> xh
  gn_stats<<<B * 16, 256, 0, stream>>>(t1, stats, B);
  gn_apply_relu_add<<<tgrid((long)B * 64 * 576), 256, 0, stream>>>(t1, stats, gn_w, gn_b, h2, xh, B);
  // 10) GST qkv 64->192
  gemm1x1_wmma<<<wgrid((long)B * 18 * 12), 256, 0, stream>>>(
      xh, gst_qkv_w, nullptr, nullptr, r3, B, 64, 64, 192, 192, 0, 576, 0);
  // 11) depthwise 3x3 pad1 groups=192
  dw3x3<<<tgrid((long)B * 192 * 576), 256, 0, stream>>>(r3, gst_dw_w, r4, B, 192, 0);
  // 12-14) channel attention
  row_invnorm<<<B * 128, 256, 0, stream>>>(r4, invn, B);
  gst_attn<<<B * 16, 64, 0, stream>>>(r4, invn, gst_temp, att, B);
  gst_o<<<tgrid((long)B * 64 * 576), 256, 0, stream>>>(r4, att, r3, B);
  // 15) proj + residual xh -> xg
  gemm1x1_wmma<<<wgrid((long)B * 18 * 4), 256, 0, stream>>>(
      r3, gst_proj_w, nullptr, xh, xg, B, 64, 64, 64, 64, 0, 576, 0);
  // 16) LayerNorm over C -> xh (reused)
  layernorm_ch<<<tgrid((long)B * 576), 256, 0, stream>>>(xg, ln_w, ln_b, xh, B);
  // 17) FFN w1 64->256 + gelu
  gemm1x1_wmma<<<wgrid((long)B * 18 * 16), 256, 0, stream>>>(
      xh, ff_w1, nullptr, nullptr, r3, B, 64, 64, 256, 256, 0, 576, 2);
  // 18) FFN depthwise + gelu
  dw3x3<<<tgrid((long)B * 256 * 576), 256, 0, stream>>>(r3, ff_dw_w, r4, B, 256, 2);
  // 19) FFN w2 256->64 + residual xg -> x11
  gemm1x1_wmma<<<wgrid((long)B * 18 * 4), 256, 0, stream>>>(
      r4, ff_w2, nullptr, xg, x11, B, 256, 256, 64, 64, 0, 576, 0);
  // 20) fused upsample + concat + cls -> out
  final_cls<<<tgrid((long)B * 66 * 66), 256, 0, stream>>>(
      h0, h1, x11, cls_w, cls_b, (float*)d_out, B);
}